// TripletContrastiveLoss_8624294331099
// MI455X (gfx1250) — compile-verified
//
#include <hip/hip_runtime.h>

typedef __attribute__((ext_vector_type(2))) float v2f;
typedef __attribute__((ext_vector_type(4))) float v4f;
typedef __attribute__((ext_vector_type(8))) float v8f;
typedef int v4i __attribute__((vector_size(16)));

#define GLOBAL_AS __attribute__((address_space(1)))
#define LDS_AS    __attribute__((address_space(3)))

#define TL_EPS    1e-8f
#define TL_MARGIN 0.5f
#define TL_D      256
#define TL_KT     64          // K-chunk staged in LDS per double-buffer slot
#define TL_LROW   68          // padded LDS row stride (floats): banks (4r+2h) -> conflict-free
#define TL_WAVES  4           // waves per block in main kernel

#if defined(__has_builtin)
#if __has_builtin(__builtin_amdgcn_global_load_async_to_lds_b128) && \
    __has_builtin(__builtin_amdgcn_s_wait_asynccnt)
#define TL_USE_ASYNC 1
#endif
#endif
#ifndef TL_USE_ASYNC
#define TL_USE_ASYNC 0
#endif

// ---------------------------------------------------------------------------
// Kernel A: per-batch-row positive cosine sim and anchor norm. One wave / row.
// ---------------------------------------------------------------------------
__global__ void tl_precompute(const float* __restrict__ anchor,
                              const float* __restrict__ pos,
                              float* __restrict__ pos_sim,
                              float* __restrict__ anchor_norm,
                              int B) {
  const int row  = (int)((blockIdx.x * blockDim.x + threadIdx.x) >> 5);
  const int lane = threadIdx.x & 31;
  if (row >= B) return;
  const float* a = anchor + (size_t)row * TL_D;
  const float* p = pos    + (size_t)row * TL_D;
  float dot = 0.f, na2 = 0.f, np2 = 0.f;
#pragma unroll
  for (int i = lane; i < TL_D; i += 32) {
    float av = a[i], pv = p[i];
    dot = fmaf(av, pv, dot);
    na2 = fmaf(av, av, na2);
    np2 = fmaf(pv, pv, np2);
  }
#pragma unroll
  for (int m = 16; m >= 1; m >>= 1) {
    dot += __shfl_xor(dot, m, 32);
    na2 += __shfl_xor(na2, m, 32);
    np2 += __shfl_xor(np2, m, 32);
  }
  if (lane == 0) {
    float na = sqrtf(na2), np = sqrtf(np2);
    pos_sim[row]     = dot / fmaxf(na * np, TL_EPS);
    anchor_norm[row] = na;
  }
}

// ---------------------------------------------------------------------------
// Kernel B: one wave per 16-negative tile.
//   cd = WMMA(anchor_gathered, negT) -> diag = dot(anchor[idx[n]], neg[n])
//   cn = WMMA(neg,             negT) -> diag = ||neg[n]||^2
// neg tile is staged global->LDS with async-to-LDS (double buffered).
// ---------------------------------------------------------------------------
__global__ __launch_bounds__(TL_WAVES * 32) void tl_main(
    const float* __restrict__ anchor,
    const float* __restrict__ neg,
    const long long* __restrict__ nidx,
    const float* __restrict__ pos_sim,
    const float* __restrict__ anchor_norm,
    float* __restrict__ sums,
    float* __restrict__ counts,
    long long N) {
  __shared__ float smem[TL_WAVES][2][16 * TL_LROW];

  const int lane = threadIdx.x & 31;
  const int wave = threadIdx.x >> 5;
  const int r    = lane & 15;   // matrix row this lane feeds
  const int h    = lane >> 4;   // lane-half selects K offset (+2h)
  const long long n0 = (long long)(blockIdx.x * TL_WAVES + wave) * 16;

  // gathered anchor row for this lane's matrix row (clamped for tails)
  long long nr = n0 + r;
  if (nr >= N) nr = N - 1;
  const long long arowi = nidx[nr];
  const float* arow = anchor + (size_t)arowi * TL_D + 2 * h;

  float* buf0 = &smem[wave][0][0];
  float* buf1 = &smem[wave][1][0];

  // ---- stage one 16 x TL_KT chunk of neg into LDS (coalesced, padded rows)
  auto prefetch = [&](int kk, float* bp) {
#pragma unroll
    for (int i = 0; i < 8; ++i) {
      const int row = 2 * i + h;                 // 2 rows per b128 instruction
      long long grow = n0 + row;
      if (grow >= N) grow = N - 1;
      const float* gp = neg + (size_t)grow * TL_D + kk + (lane & 15) * 4;
      float* lp = bp + row * TL_LROW + (lane & 15) * 4;
#if TL_USE_ASYNC
      __builtin_amdgcn_global_load_async_to_lds_b128(
          (v4i*)gp, (LDS_AS v4i*)lp, 0, 0);
#else
      *(v4f*)lp = *(const v4f*)gp;
#endif
    }
  };

  v8f cd = {};  // dot accumulator (16x16)
  v8f cn = {};  // norm accumulator (16x16)

  prefetch(0, buf0);
#pragma unroll
  for (int c = 0; c < TL_D / TL_KT; ++c) {
    float* cur = (c & 1) ? buf1 : buf0;
    float* nxt = (c & 1) ? buf0 : buf1;
    const bool more = (c + 1) < (TL_D / TL_KT);
    if (more) prefetch((c + 1) * TL_KT, nxt);
#if TL_USE_ASYNC
    if (more) __builtin_amdgcn_s_wait_asynccnt(8);   // chunk c landed
    else      __builtin_amdgcn_s_wait_asynccnt(0);
#endif
    const float* bufp = cur + r * TL_LROW + 2 * h;
    const float* ap   = arow + c * TL_KT;
#pragma unroll
    for (int k = 0; k < TL_KT; k += 4) {
      v2f b = *(const v2f*)(bufp + k);   // neg tile: B operand == A' operand
      v2f a = *(const v2f*)(ap + k);     // gathered anchor rows: A operand
      cd = __builtin_amdgcn_wmma_f32_16x16x4_f32(false, a, false, b,
                                                 (short)0, cd, false, false);
      cn = __builtin_amdgcn_wmma_f32_16x16x4_f32(false, b, false, b,
                                                 (short)0, cn, false, false);
    }
  }

  // ---- diagonal lives lane-locally: diag(i) in lane i (i<8), lane 24+(i-8)
  float dot = 0.f, nrm2 = 0.f;
  bool active = false;
#pragma unroll
  for (int i = 0; i < 8; ++i) {
    if (lane == i)      { dot = cd[i]; nrm2 = cn[i]; active = true; }
    if (lane == 24 + i) { dot = cd[i]; nrm2 = cn[i]; active = true; }
  }
  if (active && (n0 + r) < N) {
    const float na  = anchor_norm[arowi];
    const float nn  = sqrtf(nrm2);
    const float sim = dot / fmaxf(na * nn, TL_EPS);
    const float trip = fmaxf(TL_MARGIN + sim - pos_sim[arowi], 0.f);
    atomicAdd(&sums[arowi], trip);
    atomicAdd(&counts[arowi], 1.0f);
  }
}

// ---------------------------------------------------------------------------
// Kernel C: loss = (1/B) * sum_b (counts>0 ? sums/max(counts,1) : 0)
// ---------------------------------------------------------------------------
__global__ void tl_finalize(const float* __restrict__ sums,
                            const float* __restrict__ counts,
                            float* __restrict__ out, int B) {
  __shared__ float red[256];
  float acc = 0.f;
  for (int b = threadIdx.x; b < B; b += 256) {
    float c = counts[b];
    acc += (c > 0.f) ? sums[b] / fmaxf(c, 1.f) : 0.f;
  }
  red[threadIdx.x] = acc;
  __syncthreads();
#pragma unroll
  for (int s = 128; s > 0; s >>= 1) {
    if (threadIdx.x < s) red[threadIdx.x] += red[threadIdx.x + s];
    __syncthreads();
  }
  if (threadIdx.x == 0) out[0] = red[0] / (float)B;
}

extern "C" void kernel_launch(void* const* d_in, const int* in_sizes, int n_in,
                              void* d_out, int out_size, void* d_ws, size_t ws_size,
                              hipStream_t stream) {
  const float* anchor    = (const float*)d_in[0];
  const float* pos       = (const float*)d_in[1];
  const float* neg       = (const float*)d_in[2];
  const long long* nidx  = (const long long*)d_in[3];

  const int B = in_sizes[0] / TL_D;
  const long long N = (long long)in_sizes[3];

  float* ws          = (float*)d_ws;
  float* pos_sim     = ws;
  float* anchor_norm = ws + B;
  float* sums        = ws + 2 * (size_t)B;
  float* counts      = ws + 3 * (size_t)B;

  (void)hipMemsetAsync(sums, 0, (size_t)2 * B * sizeof(float), stream);

  const int preBlocks = (B * 32 + 255) / 256;
  tl_precompute<<<preBlocks, 256, 0, stream>>>(anchor, pos, pos_sim, anchor_norm, B);

  const long long tiles  = (N + 15) / 16;
  const int blocks = (int)((tiles + TL_WAVES - 1) / TL_WAVES);
  tl_main<<<blocks, TL_WAVES * 32, 0, stream>>>(anchor, neg, nidx, pos_sim,
                                                anchor_norm, sums, counts, N);

  tl_finalize<<<1, 256, 0, stream>>>(sums, counts, (float*)d_out, B);
}